// SACConv1_8143257993619
// MI455X (gfx1250) — compile-verified
//
#include <hip/hip_runtime.h>
#include <math.h>

// ---------------------------------------------------------------------------
// SACConv forward for MI455X (gfx1250, wave32).
//
// Reformulation: proj = h . (phi_w^T @ patch) + phi_b . patch, so we compute
//   T     = patches @ phi_w          (N x 288)@(288 x 64)  -- fp32 WMMA
//   alpha = h @ alpha_w^T            (N x 64)@(64 x 64)    -- fp32 WMMA
// and never materialize the (N x 288) phi matrix.
//
// fp32 WMMA (v_wmma_f32_16x16x4_f32): the problem is ~1.7 GFLOP vs ~170 MB of
// HBM traffic -> bandwidth-bound at 23.3 TB/s, so we keep fp32 precision
// parity with the reference instead of chasing f16/fp8 matrix ceilings.
//
// CDNA5 data movement: the B panels (phi_w: 72 KB, alpha_w: 16 KB) are DMA'd
// once per workgroup into LDS with the Tensor Data Mover
// (tensor_load_to_lds + s_wait_tensorcnt), so per-K-step B fragments come
// from LDS (ds_load) and VMEM bandwidth is reserved for the streaming A side.
// ---------------------------------------------------------------------------

typedef __attribute__((ext_vector_type(2))) float v2f;
typedef __attribute__((ext_vector_type(8))) float v8f;
typedef unsigned int u32x4 __attribute__((ext_vector_type(4)));
typedef int i32x4 __attribute__((ext_vector_type(4)));
typedef int i32x8 __attribute__((ext_vector_type(8)));

#define BB    4
#define C1_   32
#define C2_   64
#define HH    96
#define WW    96
#define LL    (HH * WW)        // 9216
#define NN    (BB * LL)        // 36864
#define CKK   (C1_ * 9)        // 288
#define HD    64
#define NBINS 16

// ---- Tensor Data Mover: 2-D fp32 tile Global -> LDS (D# per ISA ch.8) -----
// dim0 = elements per row (contiguous), dim1 = rows, stride0 = row stride.
// This toolchain (clang-23 / therock-10.0 headers) uses the 6-arg builtin:
//   (u32x4 group0, i32x8 group1, i32x4 group2, i32x4 group3, i32x8, i32 cpol)
__device__ __forceinline__ void tdm_load_2d_f32(const void* gptr, unsigned lds_off,
                                                unsigned dim0, unsigned dim1,
                                                unsigned stride0) {
  unsigned long long ga = (unsigned long long)(size_t)gptr;
  u32x4 g0;
  g0[0] = 1u;                                            // count=1 user D#
  g0[1] = lds_off;                                       // lds_addr (bytes)
  g0[2] = (unsigned)(ga & 0xFFFFFFFFu);                  // global_addr[31:0]
  g0[3] = (unsigned)((ga >> 32) & 0x1FFFFFFu)            // global_addr[56:32]
          | (2u << 30);                                  // type = 2 ("image")
  i32x8 g1;
  g1[0] = (int)(2u << 16);                               // data_size=4B, mask=0
  g1[1] = (int)((dim0 & 0xFFFFu) << 16);                 // tensor_dim0[15:0]
  g1[2] = (int)((dim0 >> 16) | ((dim1 & 0xFFFFu) << 16));// dim0 hi | dim1 lo
  g1[3] = (int)((dim1 >> 16) | ((dim0 & 0xFFFFu) << 16));// dim1 hi | tile_dim0
  g1[4] = (int)(dim1 & 0xFFFFu);                         // tile_dim1, tile_dim2=0
  g1[5] = (int)stride0;                                  // tensor_dim0_stride lo
  g1[6] = 0;                                             // stride hi | d1stride lo
  g1[7] = 0;
  i32x4 z4 = {0, 0, 0, 0};                               // groups 2/3 unused (2-D)
  i32x8 z8 = {0, 0, 0, 0, 0, 0, 0, 0};
  __builtin_amdgcn_tensor_load_to_lds(g0, g1, z4, z4, z8, 0);
}

// ---------------------------------------------------------------- im2col ---
__global__ void __launch_bounds__(256) sac_im2col(const float* __restrict__ x,
                                                  float* __restrict__ patches) {
  long id = (long)blockIdx.x * 256 + threadIdx.x;
  if (id >= (long)NN * CKK) return;
  int n = (int)(id / CKK), j = (int)(id % CKK);
  int b = n / LL, l = n % LL;
  int i0 = l / WW, j0 = l % WW;
  int c = j / 9, r = j % 9, kh = r / 3, kw = r % 3;
  int yy = i0 + kh - 1, xx = j0 + kw - 1;
  float v = 0.0f;
  if ((unsigned)yy < (unsigned)HH && (unsigned)xx < (unsigned)WW)
    v = x[(((size_t)b * C1_ + c) * HH + yy) * WW + xx];
  patches[(size_t)n * CKK + j] = v;
}

// ------------------------------------------------- per-column statistics ---
// One thread per spatial location; private 16-slot LDS histogram per thread,
// laid out hist[bin*256 + tid] so a wave's lanes hit distinct banks.
__global__ void __launch_bounds__(256) sac_stats(const float* __restrict__ patches,
                                                 float* __restrict__ stats) {
  __shared__ float hist[NBINS * 256];
  const int tid = threadIdx.x;
  const int n = blockIdx.x * 256 + tid;
  const float* row = patches + (size_t)n * CKK;

  float sum = 0.f, sumsq = 0.f, pmin = 3.0e38f, pmax = -3.0e38f;
  for (int j = 0; j < CKK; ++j) {
    float v = row[j];
    sum += v; sumsq += v * v;
    pmin = fminf(pmin, v); pmax = fmaxf(pmax, v);
  }
  const float inv_ckk = 1.0f / (float)CKK;
  float mu = sum * inv_ckk;
  float var = fmaxf(sumsq * inv_ckk - mu * mu, 0.0f);
  float sigma = sqrtf(var + 1e-6f);
  float zden = 1.0f / (sigma + 1e-6f);
  float rng = 15.0f / (pmax - pmin + 1e-6f);

#pragma unroll
  for (int bi = 0; bi < NBINS; ++bi) hist[bi * 256 + tid] = 0.0f;

  float z3 = 0.f, z4 = 0.f;
  for (int j = 0; j < CKK; ++j) {
    float v = row[j];
    float z = (v - mu) * zden;
    float z2 = z * z;
    z3 += z2 * z;
    z4 += z2 * z2;
    int bi = (int)((v - pmin) * rng);
    bi = min(max(bi, 0), NBINS - 1);
    hist[bi * 256 + tid] += 1.0f;            // private slot: no atomics needed
  }

  float ent = 0.f;
  const float invc = 1.0f / ((float)CKK + 1e-6f);
#pragma unroll
  for (int bi = 0; bi < NBINS; ++bi) {
    float p = hist[bi * 256 + tid] * invc;
    ent -= p * __logf(p + 1e-9f);
  }

  float* s = stats + (size_t)n * 8;          // padded to 8 for alignment
  s[0] = mu;
  s[1] = sigma;
  s[2] = z3 * inv_ckk;
  s[3] = z4 * inv_ckk - 3.0f;
  s[4] = ent;
}

// ------------------------------------------------------- h = relu(s@W+b) ---
__global__ void __launch_bounds__(256) sac_fc1(const float* __restrict__ stats,
                                               const float* __restrict__ fc1_w,
                                               const float* __restrict__ fc1_b,
                                               float* __restrict__ h) {
  long id = (long)blockIdx.x * 256 + threadIdx.x;
  if (id >= (long)NN * HD) return;
  int n = (int)(id >> 6), k = (int)(id & 63);
  const float* s = stats + (size_t)n * 8;
  const float* w = fc1_w + k * 5;
  float acc = fc1_b[k];
#pragma unroll
  for (int i = 0; i < 5; ++i) acc += s[i] * w[i];
  h[id] = fmaxf(acc, 0.0f);
}

// ---------------------------------------- T = patches @ phi_w (fp32 WMMA) ---
// Block = 128 threads = 4 waves; each wave owns one 16-row M tile and all
// four 16-wide N tiles (full C=64).  K = 288 -> 72 steps x 4 wmma.
// phi_w (288x64 fp32 = 72 KB) is TDM-DMA'd to dynamic LDS once per block.
__global__ void __launch_bounds__(128) sac_gemmT(const float* __restrict__ patches,
                                                 const float* __restrict__ phi_w,
                                                 float* __restrict__ T) {
  extern __shared__ float ldsB[];   // CKK * 64 floats (73,728 B)
  const int lane = threadIdx.x & 31;
  const int wave = threadIdx.x >> 5;

  if (wave == 0) {                  // one TDM op per workgroup (EXEC ignored)
    tdm_load_2d_f32(phi_w, (unsigned)(size_t)&ldsB[0], C2_, CKK, C2_);
    __builtin_amdgcn_s_wait_tensorcnt(0);
  }
  __syncthreads();

  const int n0 = (blockIdx.x * 4 + wave) * 16;
  const int m = lane & 15;          // A row / B column within tile
  const int khalf = lane >> 4;      // lanes 16-31 hold K=2,3 of each fragment

  const float* Arow = patches + (size_t)(n0 + m) * CKK;
  __builtin_prefetch(Arow, 0, 1);   // global_prefetch_b8: warm the A row
  v8f acc[4] = {v8f{}, v8f{}, v8f{}, v8f{}};

  for (int k0 = 0; k0 < CKK; k0 += 4) {
    // A 16x4 fragment: 2 consecutive K values per lane (b64 load)
    v2f a = *(const v2f*)(Arow + k0 + 2 * khalf);
    // B 4x16 fragments from LDS: VGPR0 = row (k0+2*khalf), VGPR1 = next row
    const float* B0 = ldsB + (size_t)(k0 + 2 * khalf) * C2_;
#pragma unroll
    for (int t = 0; t < 4; ++t) {
      v2f bf;
      bf.x = B0[t * 16 + m];
      bf.y = B0[C2_ + t * 16 + m];
      acc[t] = __builtin_amdgcn_wmma_f32_16x16x4_f32(
          false, a, false, bf, (short)0, acc[t], false, false);
    }
  }

  // C/D layout: VGPR r holds row (r + 8*khalf), column = lane&15 per N tile
#pragma unroll
  for (int r = 0; r < 8; ++r) {
    float* trow = T + (size_t)(n0 + r + 8 * khalf) * C2_;
#pragma unroll
    for (int t = 0; t < 4; ++t) trow[t * 16 + m] = acc[t][r];
  }
}

// -------------------------------- proj = h.T_row + phi_b . patch (per n) ---
__global__ void __launch_bounds__(256) sac_proj(const float* __restrict__ h,
                                                const float* __restrict__ T,
                                                const float* __restrict__ patches,
                                                const float* __restrict__ phi_b,
                                                float* __restrict__ proj) {
  int n = blockIdx.x * 256 + threadIdx.x;
  if (n >= NN) return;
  const float* hr = h + (size_t)n * HD;
  const float* tr = T + (size_t)n * HD;
  float p = 0.f;
#pragma unroll 8
  for (int k = 0; k < HD; ++k) p += hr[k] * tr[k];
  const float* pr = patches + (size_t)n * CKK;
#pragma unroll 4
  for (int j = 0; j < CKK; ++j) p += phi_b[j] * pr[j];
  proj[n] = p;
}

// ------------- alpha = h @ alpha_w^T (fp32 WMMA) + BN + SiLU + transpose ---
// alpha_w (64x64 fp32 = 16 KB) is TDM-DMA'd to LDS once per block.
__global__ void __launch_bounds__(128) sac_out(const float* __restrict__ h,
                                               const float* __restrict__ alpha_w,
                                               const float* __restrict__ alpha_b,
                                               const float* __restrict__ proj,
                                               const float* __restrict__ bn_gamma,
                                               const float* __restrict__ bn_beta,
                                               const float* __restrict__ bn_mean,
                                               const float* __restrict__ bn_var,
                                               float* __restrict__ out) {
  __shared__ float ldsA[HD * HD];   // alpha_w staged row-major (16 KB)
  const int lane = threadIdx.x & 31;
  const int wave = threadIdx.x >> 5;

  if (wave == 0) {
    tdm_load_2d_f32(alpha_w, (unsigned)(size_t)&ldsA[0], HD, C2_, HD);
    __builtin_amdgcn_s_wait_tensorcnt(0);
  }
  __syncthreads();

  const int n0 = (blockIdx.x * 4 + wave) * 16;
  const int m = lane & 15;
  const int khalf = lane >> 4;

  const float* Arow = h + (size_t)(n0 + m) * HD;
  v8f acc[4] = {v8f{}, v8f{}, v8f{}, v8f{}};

  for (int k0 = 0; k0 < HD; k0 += 4) {
    v2f a = *(const v2f*)(Arow + k0 + 2 * khalf);
    const int kr = k0 + 2 * khalf;    // even -> 8B-aligned ds b64 loads below
#pragma unroll
    for (int t = 0; t < 4; ++t) {
      // B[k][c] = alpha_w[c*64 + k]; rows kr and kr+1 adjacent in memory
      v2f bf = *(const v2f*)(ldsA + (size_t)(t * 16 + m) * HD + kr);
      acc[t] = __builtin_amdgcn_wmma_f32_16x16x4_f32(
          false, a, false, bf, (short)0, acc[t], false, false);
    }
  }

  const int b = n0 / LL;
  const int l0 = n0 % LL;           // tiles never straddle a batch (9216%16==0)
  float pr[8];
#pragma unroll
  for (int r = 0; r < 8; ++r) pr[r] = proj[n0 + 8 * khalf + r];

#pragma unroll
  for (int t = 0; t < 4; ++t) {
    const int c = t * 16 + m;
    const float inv = bn_gamma[c] * rsqrtf(bn_var[c] + 1e-5f);
    const float ab = alpha_b[c], mn = bn_mean[c], bt = bn_beta[c];
    float* orow = out + ((size_t)(b * C2_ + c)) * LL + l0 + 8 * khalf;
#pragma unroll
    for (int r = 0; r < 8; ++r) {
      float al = acc[t][r] + ab;
      float o = al * pr[r];
      float y = (o - mn) * inv + bt;
      orow[r] = y * (1.0f / (1.0f + __expf(-y)));   // SiLU; 8 contiguous floats
    }
  }
}

// ---------------------------------------------------------------- launch ---
extern "C" void kernel_launch(void* const* d_in, const int* in_sizes, int n_in,
                              void* d_out, int out_size, void* d_ws, size_t ws_size,
                              hipStream_t stream) {
  const float* x        = (const float*)d_in[0];
  const float* fc1_w    = (const float*)d_in[1];
  const float* fc1_b    = (const float*)d_in[2];
  const float* alpha_w  = (const float*)d_in[3];
  const float* alpha_b  = (const float*)d_in[4];
  const float* phi_w    = (const float*)d_in[5];
  const float* phi_b    = (const float*)d_in[6];
  const float* bn_gamma = (const float*)d_in[7];
  const float* bn_beta  = (const float*)d_in[8];
  const float* bn_mean  = (const float*)d_in[9];
  const float* bn_var   = (const float*)d_in[10];
  float* out = (float*)d_out;

  // workspace layout (floats): patches | stats(pad8) | h | T | proj  ~63 MB
  float* ws      = (float*)d_ws;
  float* patches = ws;                                   // NN*CKK
  float* stats   = patches + (size_t)NN * CKK;           // NN*8
  float* h       = stats + (size_t)NN * 8;               // NN*64
  float* T       = h + (size_t)NN * HD;                  // NN*64
  float* proj    = T + (size_t)NN * HD;                  // NN

  (void)in_sizes; (void)n_in; (void)out_size; (void)ws_size;

  const long n_im2col = (long)NN * CKK;
  sac_im2col<<<dim3((unsigned)((n_im2col + 255) / 256)), dim3(256), 0, stream>>>(x, patches);
  sac_stats<<<dim3(NN / 256), dim3(256), 0, stream>>>(patches, stats);
  const long n_fc1 = (long)NN * HD;
  sac_fc1<<<dim3((unsigned)((n_fc1 + 255) / 256)), dim3(256), 0, stream>>>(stats, fc1_w, fc1_b, h);
  sac_gemmT<<<dim3(NN / 64), dim3(128), (unsigned)(CKK * C2_ * sizeof(float)), stream>>>(
      patches, phi_w, T);                                 // 576 blocks x 4 waves, 72KB LDS
  sac_proj<<<dim3(NN / 256), dim3(256), 0, stream>>>(h, T, patches, phi_b, proj);
  sac_out<<<dim3(NN / 64), dim3(128), 0, stream>>>(h, alpha_w, alpha_b, proj,
                                                   bn_gamma, bn_beta, bn_mean, bn_var, out);
}